// EFGPND_78254304133825
// MI455X (gfx1250) — compile-verified
//
#include <hip/hip_runtime.h>
#include <hip/hip_bf16.h>

// ---------------- problem constants ----------------
#define MH      768
#define MN      1537        // 2*MH+1 nodes
#define TWO_M   1536
#define LF      3073        // 4*MH+1
#define MP      1552        // padded to 97*16
#define NCH     388         // MP/4 K-chunks for wmma f32 16x16x4
#define NPRB    64
#define NSTEP   50
#define CGIT    50
#define NTRAIN  16384
#define BTEST   4096
#define HSP     0.01f
#define SIGSQ   0.1f
#define ELLP    0.1f
#define TWOPI_F 6.28318530717958647692f
#define PI_F    3.14159265358979323846f

#define SL_RS     0
#define SL_PAP    1
#define SL_RSNEW  2
#define SL_LOGDET 3

typedef float v2f __attribute__((ext_vector_type(2)));
typedef float v8f __attribute__((ext_vector_type(8)));

// ---------------- helpers ----------------
__device__ __forceinline__ float blockReduceSum(float v, volatile float* sb) {
  __syncthreads();
  sb[threadIdx.x] = v;
  __syncthreads();
  for (int s = blockDim.x >> 1; s > 0; s >>= 1) {
    if ((int)threadIdx.x < s) sb[threadIdx.x] += sb[threadIdx.x + s];
    __syncthreads();
  }
  return sb[0];
}

// zero a float region (used to clear V padding so padded-K garbage is benign)
__global__ void k_zero(float* p, int n) {
  int i = blockIdx.x * blockDim.x + threadIdx.x;
  if (i < n) p[i] = 0.f;
}

// ---------------- stage 1: NUFFT sums ----------------
// V(q) = sum_n exp(i*2*pi*h*x_n*q), q in [-2m,2m]; compute q>=0, use conj symmetry.
// VreSh/VimSh point 16 floats into a zero-padded allocation.
__global__ void k_vq(const float* __restrict__ x, float* VreSh, float* VimSh) {
  __shared__ float sb[256];
  const float q = (float)blockIdx.x;           // 0..1536
  float cs = 0.f, sn = 0.f;
  for (int n = threadIdx.x; n < NTRAIN; n += blockDim.x) {
    float ang = TWOPI_F * HSP * x[n] * q;
    float s, c; __sincosf(ang, &s, &c);
    cs += c; sn += s;
  }
  float C = blockReduceSum(cs, sb);
  float S = blockReduceSum(sn, sb);
  if (threadIdx.x == 0) {
    int qi = blockIdx.x;
    VreSh[TWO_M + qi] = C;  VimSh[TWO_M + qi] = S;
    VreSh[TWO_M - qi] = C;  VimSh[TWO_M - qi] = -S;
  }
}

// ws[k] = sqrt(h * sqrt(2pi)*ell*exp(-2*(pi*ell*h*(k-m))^2)), zero-padded to MP
__global__ void k_ws(float* ws) {
  int i = blockIdx.x * blockDim.x + threadIdx.x;
  if (i >= MP) return;
  if (i < MN) {
    float xi = HSP * (float)(i - MH);
    float S = 2.50662827463100050242f * ELLP * __expf(-2.f * PI_F * PI_F * ELLP * ELLP * xi * xi);
    ws[i] = sqrtf(HSP * S);
  } else {
    ws[i] = 0.f;
  }
}

// rhs[k] = ws[k] * sum_n y_n * exp(-i*2*pi*h*x_n*(k-m))
__global__ void k_rhs(const float* __restrict__ x, const float* __restrict__ y,
                      const float* __restrict__ ws, float* rr, float* ri) {
  __shared__ float sb[256];
  const float f = (float)((int)blockIdx.x - MH);
  float cr = 0.f, ci = 0.f;
  for (int n = threadIdx.x; n < NTRAIN; n += blockDim.x) {
    float ang = TWOPI_F * HSP * x[n] * f;
    float s, c; __sincosf(ang, &s, &c);
    cr += y[n] * c; ci -= y[n] * s;
  }
  float CR = blockReduceSum(cr, sb);
  float CI = blockReduceSum(ci, sb);
  if (threadIdx.x == 0) { rr[blockIdx.x] = ws[blockIdx.x] * CR; ri[blockIdx.x] = ws[blockIdx.x] * CI; }
}

// ---------------- CG for beta ----------------
__global__ void k_cginit(const float* rhsr, const float* rhsi,
                         float* xr, float* xi, float* rr, float* ri,
                         float* pr, float* pi, float* apr, float* api) {
  int i = blockIdx.x * blockDim.x + threadIdx.x;
  if (i >= MP) return;
  float br = (i < MN) ? rhsr[i] : 0.f;
  float bi = (i < MN) ? rhsi[i] : 0.f;
  xr[i] = 0.f; xi[i] = 0.f;
  rr[i] = br;  ri[i] = bi;
  pr[i] = br;  pi[i] = bi;
  apr[i] = 0.f; api[i] = 0.f;
}

// Ap = ws .* (T*(ws.*p)) + sigma^2*p, T[i,j] = V(j-i)
__global__ void k_amean(const float* __restrict__ VreSh, const float* __restrict__ VimSh,
                        const float* __restrict__ ws,
                        const float* __restrict__ pr, const float* __restrict__ pi,
                        float* apr, float* api) {
  __shared__ float sb[256];
  const int i = blockIdx.x;                   // 0..MN-1
  float ar = 0.f, ai = 0.f;
  for (int j = threadIdx.x; j < MN; j += blockDim.x) {
    float tr = VreSh[j - i + TWO_M], ti = VimSh[j - i + TWO_M];
    float ur = ws[j] * pr[j], ui = ws[j] * pi[j];
    ar += tr * ur - ti * ui;
    ai += tr * ui + ti * ur;
  }
  float AR = blockReduceSum(ar, sb);
  float AI = blockReduceSum(ai, sb);
  if (threadIdx.x == 0) {
    apr[i] = ws[i] * AR + SIGSQ * pr[i];
    api[i] = ws[i] * AI + SIGSQ * pi[i];
  }
}

// slot = Re(a^H b), single block
__global__ void k_dotc(const float* ar, const float* ai, const float* br, const float* bi,
                       int n, float* slot) {
  __shared__ float sb[512];
  float s = 0.f;
  for (int i = threadIdx.x; i < n; i += blockDim.x)
    s += ar[i] * br[i] + ai[i] * bi[i];
  float S = blockReduceSum(s, sb);
  if (threadIdx.x == 0) *slot = S;
}

__global__ void k_cgupd1(const float* slots, float* xr, float* xi, float* rr, float* ri,
                         const float* pr, const float* pi, const float* apr, const float* api) {
  int i = blockIdx.x * blockDim.x + threadIdx.x;
  if (i >= MP) return;
  float a = slots[SL_RS] / slots[SL_PAP];
  xr[i] += a * pr[i];  xi[i] += a * pi[i];
  rr[i] -= a * apr[i]; ri[i] -= a * api[i];
}

__global__ void k_cgupd2(const float* slots, float* pr, float* pi,
                         const float* rr, const float* ri) {
  int i = blockIdx.x * blockDim.x + threadIdx.x;
  if (i >= MP) return;
  float b = slots[SL_RSNEW] / slots[SL_RS];
  pr[i] = rr[i] + b * pr[i];
  pi[i] = ri[i] + b * pi[i];
}

__global__ void k_copyslot(float* slots) { slots[SL_RS] = slots[SL_RSNEW]; }

// mean[t] = Re( sum_k exp(i*2pi*h*xnew_t*(k-m)) * ws[k]*beta[k] )
__global__ void k_mean(const float* __restrict__ xnew, const float* __restrict__ ws,
                       const float* __restrict__ br, const float* __restrict__ bi,
                       float* out) {
  __shared__ float sb[128];
  const float xv = xnew[blockIdx.x];
  float acc = 0.f;
  for (int k = threadIdx.x; k < MN; k += blockDim.x) {
    float ang = TWOPI_F * HSP * xv * (float)(k - MH);
    float s, c; __sincosf(ang, &s, &c);
    acc += c * (ws[k] * br[k]) - s * (ws[k] * bi[k]);
  }
  float A = blockReduceSum(acc, sb);
  if (threadIdx.x == 0) out[blockIdx.x] = A;
}

// ---------------- Lanczos (batched over 64 probes) ----------------
__global__ void k_lznz2(const float* __restrict__ z, float* nz2) {
  __shared__ float sb[256];
  const int p = blockIdx.x;
  float s = 0.f;
  for (int k = threadIdx.x; k < MN; k += blockDim.x) {
    float zv = z[p * MN + k];
    float sg = (zv > 0.f) ? 1.f : ((zv < 0.f) ? -1.f : 0.f);
    s += sg * sg;
  }
  float S = blockReduceSum(s, sb);
  if (threadIdx.x == 0) nz2[p] = S;
}

// WQ2 layout: pair-interleaved for b64 B-fragment loads:
//   WQ2[(k>>1)*128 + 2*p + (k&1)] = ws[k]*Q[k][p]
__global__ void k_lzinit(const float* __restrict__ z, const float* __restrict__ nz2,
                         const float* __restrict__ ws,
                         float* Qr, float* Qi, float* Qpr, float* Qpi,
                         float* WQ2r, float* WQ2i, float* bprev) {
  int idx = blockIdx.x * blockDim.x + threadIdx.x;
  if (idx >= MP * NPRB) return;
  int k = idx >> 6, p = idx & 63;
  float q = 0.f;
  if (k < MN) {
    float zv = z[p * MN + k];
    float sg = (zv > 0.f) ? 1.f : ((zv < 0.f) ? -1.f : 0.f);
    q = sg / sqrtf(nz2[p]);
  }
  Qr[idx] = q;  Qi[idx] = 0.f;
  Qpr[idx] = 0.f; Qpi[idx] = 0.f;
  int pidx = (k >> 1) * 128 + (p << 1) + (k & 1);
  WQ2r[pidx] = ws[k] * q; WQ2i[pidx] = 0.f;
  if (k == 0) bprev[p] = 0.f;
}

// W = Q + (1/sigma^2) * ws .* (T * (ws .* Q)) - bprev(col) .* Qprev
// Complex GEMM with V_WMMA_F32_16X16X4_F32.
// Block = one 16-row M-tile, 512 threads = 16 waves = 4 N-tiles x 4 K-splits.
// No bounds checks: padded K columns have WQ == 0 (B operand zero), padded rows
// are killed by ws[mr]==0 in the epilogue; V is gathered from a zero-padded array.
__global__ void __launch_bounds__(512) k_lanczos_gemm(
    const float* __restrict__ VreSh, const float* __restrict__ VimSh,
    const float* __restrict__ ws,
    const float* __restrict__ Qr, const float* __restrict__ Qi,
    const float* __restrict__ Qpr, const float* __restrict__ Qpi,
    const float* __restrict__ WQ2r, const float* __restrict__ WQ2i,
    const float* __restrict__ bprev,
    float* __restrict__ Wr, float* __restrict__ Wi) {
  __shared__ float sRr[16 * 256];
  __shared__ float sRi[16 * 256];
  const int lane  = threadIdx.x & 31;
  const int wave  = threadIdx.x >> 5;
  const int ntile = wave & 3;
  const int ksp   = wave >> 2;
  const int m0 = blockIdx.x * 16;
  const int n0 = ntile * 16;
  const int half = lane >> 4;
  const int l15  = lane & 15;
  const int mrow_a = m0 + l15;       // A-fragment row for this lane
  const int col = n0 + l15;          // B/D column (probe)

  // A[m][k] = T[m][k] = V(k-m); per-lane shifted base pointers (always in-bounds
  // thanks to 16-float zero padding around V).
  const float* vr = VreSh + (TWO_M - mrow_a);
  const float* vi = VimSh + (TWO_M - mrow_a);
  // B fragments: pair-interleaved WQ2, one aligned v2f per K-pair
  const v2f* bqr = (const v2f*)WQ2r;
  const v2f* bqi = (const v2f*)WQ2i;

  v8f accRr = {0.f, 0.f, 0.f, 0.f, 0.f, 0.f, 0.f, 0.f};
  v8f accRi = {0.f, 0.f, 0.f, 0.f, 0.f, 0.f, 0.f, 0.f};

#pragma unroll 2
  for (int kc = ksp; kc < NCH; kc += 4) {
    const int kb = kc * 4 + half * 2;     // K index for A cols / B rows (even)
    v2f ar, ai, br, bi, nbi;
    ar.x = vr[kb];  ar.y = vr[kb + 1];
    ai.x = vi[kb];  ai.y = vi[kb + 1];
    const int qb = (kb >> 1) * 64 + col;  // v2f element index
    br = bqr[qb];
    bi = bqi[qb];
    nbi.x = -bi.x;  nbi.y = -bi.y;        // f32 WMMA has no A/B NEG bits
    // Dr += Tr*Br - Ti*Bi ; Di += Tr*Bi + Ti*Br
    accRr = __builtin_amdgcn_wmma_f32_16x16x4_f32(false, ar, false, br,  (short)0, accRr, false, false);
    accRr = __builtin_amdgcn_wmma_f32_16x16x4_f32(false, ai, false, nbi, (short)0, accRr, false, false);
    accRi = __builtin_amdgcn_wmma_f32_16x16x4_f32(false, ar, false, bi,  (short)0, accRi, false, false);
    accRi = __builtin_amdgcn_wmma_f32_16x16x4_f32(false, ai, false, br,  (short)0, accRi, false, false);
  }

  // K-split reduction through LDS
#pragma unroll
  for (int r = 0; r < 8; ++r) {
    sRr[wave * 256 + r * 32 + lane] = accRr[r];
    sRi[wave * 256 + r * 32 + lane] = accRi[r];
  }
  __syncthreads();

  if (ksp == 0) {
    const float bp = bprev[col];
    const float invs2 = 1.0f / SIGSQ;
#pragma unroll
    for (int r = 0; r < 8; ++r) {
      float vR = accRr[r], vI = accRi[r];
#pragma unroll
      for (int ks = 1; ks < 4; ++ks) {
        vR += sRr[(ks * 4 + ntile) * 256 + r * 32 + lane];
        vI += sRi[(ks * 4 + ntile) * 256 + r * 32 + lane];
      }
      const int mr = m0 + r + 8 * half;   // C/D layout: VGPR r -> M = r + 8*half
      const int idx = mr * NPRB + col;
      const float wsm = ws[mr];           // zero for padded rows -> W rows stay 0
      Wr[idx] = Qr[idx] + invs2 * wsm * vR - bp * Qpr[idx];
      Wi[idx] = Qi[idx] + invs2 * wsm * vI - bp * Qpi[idx];
    }
  }
}

// alpha_p = Re(q^H v)
__global__ void k_alpha(const float* Qr, const float* Qi, const float* Wr, const float* Wi,
                        float* alphas, float* acur, int step) {
  __shared__ float sb[256];
  const int p = blockIdx.x;
  float s = 0.f;
  for (int k = threadIdx.x; k < MP; k += blockDim.x) {
    int idx = k * NPRB + p;
    s += Qr[idx] * Wr[idx] + Qi[idx] * Wi[idx];
  }
  float A = blockReduceSum(s, sb);
  if (threadIdx.x == 0) { alphas[p * NSTEP + step] = A; acur[p] = A; }
}

// v <- v - alpha*q ; beta = ||v||
__global__ void k_vupd(const float* Qr, const float* Qi, float* Wr, float* Wi,
                       const float* acur, float* betas, float* bcur, float* bprev, int step) {
  __shared__ float sb[256];
  const int p = blockIdx.x;
  const float a = acur[p];
  float s = 0.f;
  for (int k = threadIdx.x; k < MP; k += blockDim.x) {
    int idx = k * NPRB + p;
    float vrv = Wr[idx] - a * Qr[idx];
    float viv = Wi[idx] - a * Qi[idx];
    Wr[idx] = vrv; Wi[idx] = viv;
    s += vrv * vrv + viv * viv;
  }
  float S = blockReduceSum(s, sb);
  if (threadIdx.x == 0) {
    float b = sqrtf(S);
    betas[p * NSTEP + step] = b;
    bcur[p] = b;
    bprev[p] = b;   // consumed by NEXT step's GEMM
  }
}

// qprev <- q ; q <- v / max(beta,1e-12) ; WQ2 <- ws .* q (pair-interleaved)
__global__ void k_qshift(float* Qr, float* Qi, float* Qpr, float* Qpi,
                         float* WQ2r, float* WQ2i,
                         const float* Wr, const float* Wi,
                         const float* __restrict__ ws, const float* bcur) {
  int idx = blockIdx.x * blockDim.x + threadIdx.x;
  if (idx >= MP * NPRB) return;
  int k = idx >> 6, p = idx & 63;
  float inv = 1.0f / fmaxf(bcur[p], 1e-12f);
  float qr = Qr[idx], qi = Qi[idx];
  Qpr[idx] = qr; Qpi[idx] = qi;
  float nqr = Wr[idx] * inv;
  float nqi = Wi[idx] * inv;
  Qr[idx] = nqr; Qi[idx] = nqi;
  float w = ws[k];
  int pidx = (k >> 1) * 128 + (p << 1) + (k & 1);
  WQ2r[pidx] = w * nqr; WQ2i[pidx] = w * nqi;
}

// ---------------- SLQ: tridiag eig (imtql2 w/ first-row update), one probe/thread ----------------
__global__ void k_slq(const float* alphas, const float* betas, const float* nz2,
                      float* slot_logdet) {
  __shared__ float sb[64];
  const int p = threadIdx.x;
  float d[NSTEP], e[NSTEP], zr[NSTEP];
  for (int j = 0; j < NSTEP; ++j) {
    d[j] = alphas[p * NSTEP + j];
    e[j] = (j < NSTEP - 1) ? betas[p * NSTEP + j] : 0.f;
    zr[j] = 0.f;
  }
  zr[0] = 1.f;
  const int n = NSTEP;
  for (int l = 0; l < n; ++l) {
    for (int iter = 0; iter < 40; ++iter) {
      int mm;
      for (mm = l; mm < n - 1; ++mm) {
        float dd = fabsf(d[mm]) + fabsf(d[mm + 1]);
        if (fabsf(e[mm]) <= 1.19209290e-7f * dd) break;
      }
      if (mm == l) break;
      float g = (d[l + 1] - d[l]) / (2.0f * e[l]);
      float r = sqrtf(g * g + 1.0f);
      g = d[mm] - d[l] + e[l] / (g + copysignf(r, g));
      float s = 1.0f, c = 1.0f, pp = 0.0f;
      for (int i = mm - 1; i >= l; --i) {
        float f = s * e[i], b = c * e[i];
        r = sqrtf(f * f + g * g);
        e[i + 1] = r;
        if (r == 0.0f) { d[i + 1] -= pp; e[mm] = 0.0f; break; }
        s = f / r; c = g / r;
        g = d[i + 1] - pp;
        r = (d[i] - g) * s + 2.0f * c * b;
        pp = s * r;
        d[i + 1] = g + pp;
        g = c * r - b;
        f = zr[i + 1];
        zr[i + 1] = s * zr[i] + c * f;
        zr[i]     = c * zr[i] - s * f;
      }
      d[l] -= pp; e[l] = g; e[mm] = 0.0f;
    }
  }
  float quad = 0.f;
  for (int j = 0; j < n; ++j)
    quad += zr[j] * zr[j] * logf(fmaxf(d[j], 1e-18f));
  quad *= nz2[p];
  float S = blockReduceSum(quad, sb);
  if (threadIdx.x == 0)
    *slot_logdet = S / (float)NPRB + (float)NTRAIN * logf(SIGSQ);
}

// ---------------- lml ----------------
__global__ void k_lml1(const float* __restrict__ x, const float* __restrict__ y,
                       const float* __restrict__ ws,
                       const float* __restrict__ br, const float* __restrict__ bi,
                       float* partials) {
  __shared__ float sb[256];
  const int gid = blockIdx.x * blockDim.x + threadIdx.x;
  float t = 0.f;
  if (gid < NTRAIN) {
    const float xv = x[gid], yv = y[gid];
    float s = 0.f;
    for (int k = 0; k < MN; ++k) {
      float ang = TWOPI_F * HSP * xv * (float)(k - MH);
      float sn, cs; __sincosf(ang, &sn, &cs);
      s += cs * (ws[k] * br[k]) - sn * (ws[k] * bi[k]);
    }
    t = yv * (yv - s) * (1.0f / SIGSQ);
  }
  float S = blockReduceSum(t, sb);
  if (threadIdx.x == 0) partials[blockIdx.x] = S;
}

__global__ void k_lml2(const float* partials, const float* slot_logdet, float* out_lml) {
  __shared__ float sb[64];
  float S = blockReduceSum(partials[threadIdx.x], sb);
  if (threadIdx.x == 0)
    *out_lml = -0.5f * S - 0.5f * (*slot_logdet)
               - 0.5f * (float)NTRAIN * logf(TWOPI_F);
}

// ---------------- host orchestration ----------------
extern "C" void kernel_launch(void* const* d_in, const int* in_sizes, int n_in,
                              void* d_out, int out_size, void* d_ws, size_t ws_size,
                              hipStream_t stream) {
  const float* x    = (const float*)d_in[0];   // 16384
  const float* y    = (const float*)d_in[1];   // 16384
  const float* xnew = (const float*)d_in[2];   // 4096
  const float* z    = (const float*)d_in[3];   // 64*1537
  float* out = (float*)d_out;                  // 4096 mean + 1 lml

  float* W = (float*)d_ws;
  // V arrays: 16-float front pad + 3073 + slack, regions [0,4096) and [4096,8192)
  float* VreSh = W + 16;          // shifted base; storage W[0..4095]
  float* VimSh = W + 4096 + 16;   // storage W[4096..8191]
  float* ws    = W + 8192;        // 1552
  float* rhsr  = W + 10240;
  float* rhsi  = W + 12288;
  float* xr_   = W + 14336;       // beta (re)
  float* xi_   = W + 16384;       // beta (im)
  float* rr_   = W + 18432;
  float* ri_   = W + 20480;
  float* pr_   = W + 22528;
  float* pi_   = W + 24576;
  float* apr_  = W + 26624;
  float* api_  = W + 28672;
  float* slots = W + 30720;       // scalar slots
  float* parts = W + 30784;       // 64 partials
  float* nz2   = W + 30848;       // 64
  float* acur  = W + 30912;       // 64
  float* bcur  = W + 30976;       // 64
  float* bprev = W + 31040;       // 64
  float* alphas= W + 31104;       // 64*50
  float* betas = W + 34304;       // 64*50
  float* Qr    = W + 38912;       // [MP][64] each below
  float* Qi    = Qr  + MP * NPRB;
  float* Qpr   = Qi  + MP * NPRB;
  float* Qpi   = Qpr + MP * NPRB;
  float* Wr    = Qpi + MP * NPRB;
  float* Wi    = Wr  + MP * NPRB;
  float* WQ2r  = Wi  + MP * NPRB; // pair-interleaved [MP/2][64][2]
  float* WQ2i  = WQ2r + MP * NPRB;

  const int gMP = (MP + 255) / 256;
  const int gQ  = (MP * NPRB + 255) / 256;

  // clear V storage (incl. padding), then NUFFT generator + weights + rhs
  k_zero<<<(8192 + 255) / 256, 256, 0, stream>>>(W, 8192);
  k_vq <<<MN, 256, 0, stream>>>(x, VreSh, VimSh);
  k_ws <<<gMP, 256, 0, stream>>>(ws);
  k_rhs<<<MN, 256, 0, stream>>>(x, y, ws, rhsr, rhsi);

  // CG for beta
  k_cginit<<<gMP, 256, 0, stream>>>(rhsr, rhsi, xr_, xi_, rr_, ri_, pr_, pi_, apr_, api_);
  k_dotc<<<1, 512, 0, stream>>>(rr_, ri_, rr_, ri_, MP, slots + SL_RS);
  for (int it = 0; it < CGIT; ++it) {
    k_amean<<<MN, 256, 0, stream>>>(VreSh, VimSh, ws, pr_, pi_, apr_, api_);
    k_dotc <<<1, 512, 0, stream>>>(pr_, pi_, apr_, api_, MP, slots + SL_PAP);
    k_cgupd1<<<gMP, 256, 0, stream>>>(slots, xr_, xi_, rr_, ri_, pr_, pi_, apr_, api_);
    k_dotc <<<1, 512, 0, stream>>>(rr_, ri_, rr_, ri_, MP, slots + SL_RSNEW);
    k_cgupd2<<<gMP, 256, 0, stream>>>(slots, pr_, pi_, rr_, ri_);
    k_copyslot<<<1, 1, 0, stream>>>(slots);
  }

  // posterior mean
  k_mean<<<BTEST, 128, 0, stream>>>(xnew, ws, xr_, xi_, out);

  // Lanczos / SLQ (WMMA GEMM per step)
  k_lznz2 <<<NPRB, 256, 0, stream>>>(z, nz2);
  k_lzinit<<<gQ, 256, 0, stream>>>(z, nz2, ws, Qr, Qi, Qpr, Qpi, WQ2r, WQ2i, bprev);
  for (int s = 0; s < NSTEP; ++s) {
    k_lanczos_gemm<<<MP / 16, 512, 0, stream>>>(VreSh, VimSh, ws, Qr, Qi, Qpr, Qpi,
                                                WQ2r, WQ2i, bprev, Wr, Wi);
    k_alpha <<<NPRB, 256, 0, stream>>>(Qr, Qi, Wr, Wi, alphas, acur, s);
    k_vupd  <<<NPRB, 256, 0, stream>>>(Qr, Qi, Wr, Wi, acur, betas, bcur, bprev, s);
    k_qshift<<<gQ, 256, 0, stream>>>(Qr, Qi, Qpr, Qpi, WQ2r, WQ2i, Wr, Wi, ws, bcur);
  }
  k_slq<<<1, 64, 0, stream>>>(alphas, betas, nz2, slots + SL_LOGDET);

  // log marginal likelihood
  k_lml1<<<NTRAIN / 256, 256, 0, stream>>>(x, y, ws, xr_, xi_, parts);
  k_lml2<<<1, 64, 0, stream>>>(parts, slots + SL_LOGDET, out + BTEST);
}